// GLSD_34265249087854
// MI455X (gfx1250) — compile-verified
//
#include <hip/hip_runtime.h>
#include <math.h>

// ---------------------------------------------------------------------------
// F-Clip post-processing for MI455X (gfx1250, wave32).
//   k1: channel activations (softmax2 == sigmoid(x1-x0), sigmoids)   [HBM-bound]
//   k2: 3x3 heatmap NMS                                              [HBM-bound]
//   k3: per-image exact top-k via full bitonic sort of 16384 u64 keys
//       entirely in LDS (128 KB dyn-LDS; CDNA5 WGP has 320 KB) + line build
//   k4: structure NMS as two K=4 Gram GEMMs on V_WMMA_F32_16X16X4_F32
//       dist(i,j) = nu_i + nu_j - 2*max(U.U^T, U.V^T)  (exact f32)
// ---------------------------------------------------------------------------

#define NBATCH 32
#define HW     128
#define NPIX   (HW*HW)      // 16384
#define NLN    1000         // NLINES
#define KPAD   1024         // NLN padded to 64 WMMA column tiles
#define NTILES (KPAD/16)    // 64
#define SNMS   2.0f

typedef __attribute__((ext_vector_type(2))) float v2f;
typedef __attribute__((ext_vector_type(8))) float v8f;

__device__ __forceinline__ float sigm(float x) { return 1.0f / (1.0f + expf(-x)); }

// --------------------------- kernel 1: activations -------------------------
__global__ void k_act(const float* __restrict__ in,
                      float* __restrict__ lcmap, float* __restrict__ lcoff,
                      float* __restrict__ lleng, float* __restrict__ angle)
{
    int t = blockIdx.x * blockDim.x + threadIdx.x;
    if (t >= NBATCH * NPIX) return;
    int b = t >> 14, p = t & (NPIX - 1);
    const float* base = in + (size_t)b * 6 * NPIX + p;
    float x0 = base[0 * NPIX], x1 = base[1 * NPIX], x2 = base[2 * NPIX];
    float x3 = base[3 * NPIX], x4 = base[4 * NPIX], x5 = base[5 * NPIX];
    lcmap[t] = sigm(x1 - x0);                       // softmax over 2 ch, ch 1
    lcoff[(size_t)b * 2 * NPIX + p]        = sigm(x2) - 0.5f;
    lcoff[(size_t)b * 2 * NPIX + NPIX + p] = sigm(x3) - 0.5f;
    lleng[t] = sigm(x4);
    angle[t] = sigm(x5);
}

// --------------------------- kernel 2: 3x3 heat NMS ------------------------
__global__ void k_nms(const float* __restrict__ lcmap, float* __restrict__ nmsmap)
{
    int t = blockIdx.x * blockDim.x + threadIdx.x;
    if (t >= NBATCH * NPIX) return;
    int b = t >> 14, p = t & (NPIX - 1);
    int y = p >> 7, x = p & 127;
    const float* hm = lcmap + (size_t)b * NPIX;
    float h = hm[p];
    float m = -INFINITY;
    #pragma unroll
    for (int dy = -1; dy <= 1; ++dy) {
        int yy = y + dy;
        if (yy < 0 || yy >= HW) continue;
        #pragma unroll
        for (int dx = -1; dx <= 1; ++dx) {
            int xx = x + dx;
            if (xx < 0 || xx >= HW) continue;
            m = fmaxf(m, hm[yy * HW + xx]);
        }
    }
    nmsmap[t] = (h == m) ? h : 0.0f;
}

// ---------------- kernel 3: exact top-k (bitonic in LDS) + lines -----------
// key = (~float_bits(value) << 32) | pixel_index; values are >= 0 so bits are
// monotone. Ascending u64 sort => value descending, ties by index ascending
// (matches jax.lax.top_k). One workgroup per image; 16384 keys = 128 KB LDS.
__global__ void __launch_bounds__(1024) k_topk(
    const float* __restrict__ nmsmap, const float* __restrict__ lcoff,
    const float* __restrict__ lleng,  const float* __restrict__ angle,
    float* __restrict__ lines,  // (32,1000,2,2)
    float* __restrict__ U,      // (32,1024,4)  [p1y p1x p2y p2x], padded 0
    float* __restrict__ nu,     // (32,1024)    squared norms
    float* __restrict__ rawsc)  // (32,1024)    top-k scores (pre-suppress)
{
    extern __shared__ unsigned long long keys[];
    const int b = blockIdx.x, tid = threadIdx.x;
    const float* hm = nmsmap + (size_t)b * NPIX;

    for (int i = tid; i < NPIX; i += 1024) {
        unsigned fb = __float_as_uint(hm[i]);
        keys[i] = ((unsigned long long)(~fb) << 32) | (unsigned)i;
    }
    __syncthreads();

    for (int k = 2; k <= NPIX; k <<= 1) {
        for (int j = k >> 1; j > 0; j >>= 1) {
            for (int i = tid; i < NPIX; i += 1024) {
                int ixj = i ^ j;
                if (ixj > i) {
                    unsigned long long a = keys[i], c = keys[ixj];
                    bool asc = ((i & k) == 0);
                    if ((a > c) == asc) { keys[i] = c; keys[ixj] = a; }
                }
            }
            __syncthreads();
        }
    }

    // Build line endpoints for the top 1000; zero-pad rows 1000..1023.
    if (tid < KPAD) {
        float u0 = 0.f, u1 = 0.f, u2 = 0.f, u3 = 0.f, sc = 0.f;
        if (tid < NLN) {
            unsigned long long key = keys[tid];
            unsigned p = (unsigned)(key & 0xFFFFFFFFu);
            sc = __uint_as_float(~(unsigned)(key >> 32));
            int yi = p >> 7, xi = p & 127;
            float yc = (float)yi + lcoff[(size_t)b * 2 * NPIX + p] + 0.5f;
            float xc = (float)xi + lcoff[(size_t)b * 2 * NPIX + NPIX + p] + 0.5f;
            float hl = lleng[(size_t)b * NPIX + p] * 64.0f;   // RESOLUTION/2
            float th = angle[(size_t)b * NPIX + p] * 3.14159265358979323846f;
            float dy = hl * sinf(th), dx = hl * cosf(th);
            u0 = yc - dy; u1 = xc - dx; u2 = yc + dy; u3 = xc + dx;
            float* L = lines + ((size_t)b * NLN + tid) * 4;
            L[0] = u0; L[1] = u1; L[2] = u2; L[3] = u3;
        }
        float* Ub = U + ((size_t)b * KPAD + tid) * 4;
        Ub[0] = u0; Ub[1] = u1; Ub[2] = u2; Ub[3] = u3;
        nu[(size_t)b * KPAD + tid]    = u0*u0 + u1*u1 + u2*u2 + u3*u3;
        rawsc[(size_t)b * KPAD + tid] = sc;
    }
}

// ---------------- kernel 4: structure NMS via WMMA f32 16x16x4 -------------
// One workgroup (8 wave32) per batch. U (16 KB) + norms + flags live in LDS.
// Wave w owns column tiles jt = w, w+8, ...; iterates row tiles it <= jt.
// Two WMMAs per tile pair: G1 = A * U^T, G2 = A * V^T (V = endpoint-swapped).
__global__ void __launch_bounds__(256) k_snms(
    const float* __restrict__ U, const float* __restrict__ nu,
    const float* __restrict__ rawsc, float* __restrict__ score)
{
    __shared__ float    sU[KPAD * 4];
    __shared__ float    sNu[KPAD];
    __shared__ unsigned sSupp[KPAD];
    const int b = blockIdx.x, tid = threadIdx.x;

    for (int i = tid; i < KPAD * 4; i += 256) sU[i] = U[(size_t)b * KPAD * 4 + i];
    for (int i = tid; i < KPAD; i += 256) { sNu[i] = nu[(size_t)b * KPAD + i]; sSupp[i] = 0u; }
    __syncthreads();

    const int lane = tid & 31;       // wave32
    const int wave = tid >> 5;
    const int m    = lane & 15;      // row (A) / column (B,D) within tile
    const int kh   = lane >> 4;      // K half: 0 -> K=0,1 ; 1 -> K=2,3

    for (int jt = wave; jt < NTILES; jt += 8) {
        const int col = jt * 16 + m;
        v2f bv1, bv2;                           // B = U^T and V^T tiles
        bv1.x = sU[col * 4 + 2 * kh + 0];
        bv1.y = sU[col * 4 + 2 * kh + 1];
        bv2.x = sU[col * 4 + ((2 * kh + 2) & 3)];  // v[k] = u[(k+2)&3]
        bv2.y = sU[col * 4 + ((2 * kh + 3) & 3)];
        const float nuj = sNu[col];
        unsigned flag = 0u;

        for (int it = 0; it <= jt; ++it) {      // only tiles where i<j possible
            const int row = it * 16 + m;
            v2f av;
            av.x = sU[row * 4 + 2 * kh + 0];
            av.y = sU[row * 4 + 2 * kh + 1];
            v8f c0 = {};
            union { v8f v; float f[8]; } d1, d2;
            // D = A(16x4) x B(4x16) + 0 ; exact K=4 f32 matrix op
            d1.v = __builtin_amdgcn_wmma_f32_16x16x4_f32(
                false, av, false, bv1, (short)0, c0, false, false);
            d2.v = __builtin_amdgcn_wmma_f32_16x16x4_f32(
                false, av, false, bv2, (short)0, c0, false, false);
            #pragma unroll
            for (int r = 0; r < 8; ++r) {
                int i = it * 16 + r + 8 * kh;   // D row for VGPR r, lane half
                int j = jt * 16 + m;            // D column for this lane
                float g = fmaxf(d1.f[r], d2.f[r]);
                float dist = sNu[i] + nuj - 2.0f * g;
                if (i < j && i < NLN && j < NLN && dist <= SNMS) flag = 1u;
            }
        }
        if (flag) atomicOr(&sSupp[jt * 16 + m], 1u);
    }
    __syncthreads();

    for (int j = tid; j < NLN; j += 256)
        score[(size_t)b * NLN + j] = sSupp[j] ? 0.0f : rawsc[(size_t)b * KPAD + j];
}

// ---------------------------------------------------------------------------
extern "C" void kernel_launch(void* const* d_in, const int* in_sizes, int n_in,
                              void* d_out, int out_size, void* d_ws, size_t ws_size,
                              hipStream_t stream)
{
    (void)in_sizes; (void)n_in; (void)out_size; (void)ws_size;
    const float* in = (const float*)d_in[0];

    // d_out: lcmap | lcoff | lleng | angle | lines | score (flat, f32)
    float* out   = (float*)d_out;
    float* lcmap = out;                 // 32*16384          = 524288
    float* lcoff = out + 524288;        // 32*2*16384        = 1048576
    float* lleng = out + 1572864;       // 32*16384
    float* angle = out + 2097152;       // 32*16384
    float* lines = out + 2621440;       // 32*1000*2*2       = 128000
    float* score = out + 2749440;       // 32*1000           = 32000

    // workspace: nms heat | padded point rows U | norms | raw top-k scores
    float* ws     = (float*)d_ws;
    float* nmsmap = ws;                 // 524288
    float* U      = ws + 524288;        // 32*1024*4 = 131072
    float* nu     = ws + 655360;        // 32*1024   = 32768
    float* rawsc  = ws + 688128;        // 32*1024   = 32768

    const int nPix = NBATCH * NPIX;
    k_act<<<(nPix + 255) / 256, 256, 0, stream>>>(in, lcmap, lcoff, lleng, angle);
    k_nms<<<(nPix + 255) / 256, 256, 0, stream>>>(lcmap, nmsmap);
    k_topk<<<NBATCH, 1024, NPIX * sizeof(unsigned long long), stream>>>(
        nmsmap, lcoff, lleng, angle, lines, U, nu, rawsc);
    k_snms<<<NBATCH, 256, 0, stream>>>(U, nu, rawsc, score);
}